// PeakTransmissionModel_4260607558045
// MI455X (gfx1250) — compile-verified
//
#include <hip/hip_runtime.h>
#include <math.h>

typedef __attribute__((ext_vector_type(2))) float v2f;
typedef __attribute__((ext_vector_type(8))) float v8f;

#define NPEAK 16
// TUNABLE = [0, 3, 6, 9, 12, 15] (compile-time constant in the reference)
#define TUNABLE_MASK ((1<<0)|(1<<3)|(1<<6)|(1<<9)|(1<<12)|(1<<15))
#define E_TUNE  0.1f
#define GMIN    1e-4f
#define GMAX    0.5f

// ---------------------------------------------------------------------------
// Prologue: 16 threads compute per-peak derived constants into d_ws:
//   ws[0..15]  = ec_p
//   ws[16..31] = g_p^2                (g = 0.5*(gr+gl))
//   ws[32+16*r+p] = coef(r, p), r=0..3:
//     r=0: a = num_re                      (weight of E*inv in Re)
//     r=1: c = num_im*g - num_re*ec        (weight of inv   in Re)
//     r=2: b = num_im                      (weight of E*inv in Im)
//     r=3: d = -(num_im*ec + num_re*g)     (weight of inv   in Im)
// ---------------------------------------------------------------------------
__global__ void setup_params(const float* __restrict__ ec_w, const float* __restrict__ ec_b,
                             const float* __restrict__ th_w, const float* __restrict__ th_b,
                             const float* __restrict__ gr_w, const float* __restrict__ gr_b,
                             const float* __restrict__ gl_w, const float* __restrict__ gl_b,
                             const float* __restrict__ e0,   float* __restrict__ ws) {
  int p = threadIdx.x;
  if (p >= NPEAK) return;
  float ec = ((TUNABLE_MASK >> p) & 1)
                 ? (e0[p] + E_TUNE * tanhf(ec_w[p] + ec_b[p]))
                 : e0[p];
  float th = 3.14159265358979323846f / (1.0f + expf(-(th_w[p] + th_b[p])));
  float gr = GMIN + (GMAX - GMIN) / (1.0f + expf(-(gr_w[p] + gr_b[p])));
  float gl = GMIN + (GMAX - GMIN) / (1.0f + expf(-(gl_w[p] + gl_b[p])));
  float g  = 0.5f * (gr + gl);
  float sq = sqrtf(gr * gl);
  float a  = cosf(th) * sq;   // num_re
  float b  = sinf(th) * sq;   // num_im
  ws[p]        = ec;
  ws[16 + p]   = g * g;
  ws[32 + 0*16 + p] = a;
  ws[32 + 1*16 + p] = b * g - a * ec;
  ws[32 + 2*16 + p] = b;
  ws[32 + 3*16 + p] = -(b * ec + a * g);
}

// ---------------------------------------------------------------------------
// Main kernel: each wave32 processes 16-element tiles.
//  D(16x16) = A(16x4 coef) x B(4x16 inv) accumulated over 4 K-slices (peaks).
//  Operand layouts per CDNA5 ISA (32-bit, 16x16x4):
//   A: lane L (L<16): row M=L, v[j] = A[M][2*kh + j], kh = L>>4
//   B: lane L:        col N=L&15, v[j] = B[2*kh + j][N]
//   D: lanes 0-15: acc[j] = D[M=j][N=lane]  -> element's S1..S4 in acc[0..3]
// ---------------------------------------------------------------------------
__global__ __launch_bounds__(256)
void peaks_wmma_kernel(const float* __restrict__ E, float* __restrict__ out,
                       const float* __restrict__ ws, int n) {
  const int lane = threadIdx.x & 31;
  const int m    = lane & 15;   // element-in-tile (= B's N, = A's M row)
  const int kh   = lane >> 4;   // half-wave -> K base 2*kh
  const int wave   = blockIdx.x * (blockDim.x >> 5) + (threadIdx.x >> 5);
  const int nwaves = gridDim.x * (blockDim.x >> 5);
  const int ntiles = (n + 15) >> 4;

  // Hoisted per-lane constants: ec/g2 for the 8 (slice,j) peak slots, and the
  // constant A (coefficient) operand (nonzero only in rows M<4).
  float ecv[4][2], g2v[4][2];
  v2f A[4];
#pragma unroll
  for (int s = 0; s < 4; ++s) {
#pragma unroll
    for (int j = 0; j < 2; ++j) {
      int p = 4 * s + 2 * kh + j;
      ecv[s][j] = ws[p];
      g2v[s][j] = ws[16 + p];
      A[s][j]   = (m < 4) ? ws[32 + 16 * m + p] : 0.0f;
    }
  }

  for (int t = wave; t < ntiles; t += nwaves) {
    const int base = t << 4;
    int idx = base + m;
    float Ev = E[idx < n ? idx : (n - 1)];

    // B operand: inv_{m,p} = rcp((E - ec_p)^2 + g_p^2); 8 rcps per tile.
    v2f B[4];
#pragma unroll
    for (int s = 0; s < 4; ++s) {
#pragma unroll
      for (int j = 0; j < 2; ++j) {
        float d = Ev - ecv[s][j];
        B[s][j] = __builtin_amdgcn_rcpf(__builtin_fmaf(d, d, g2v[s][j]));
      }
    }

    // Chain 4 WMMAs: accumulate over the 4 K-slices of 4 peaks each.
    v8f acc = {};
#pragma unroll
    for (int s = 0; s < 4; ++s) {
      acc = __builtin_amdgcn_wmma_f32_16x16x4_f32(
          /*neg_a=*/false, A[s], /*neg_b=*/false, B[s],
          /*c_mod=*/(short)0, acc, /*reuse_a=*/false, /*reuse_b=*/false);
    }

    // Epilogue: lanes 0-15 hold S1..S4 for element (base+lane) in acc[0..3].
    if (lane < 16 && idx < n) {
      float tre = __builtin_fmaf(Ev, acc[0], acc[1]);
      float tim = __builtin_fmaf(Ev, acc[2], acc[3]);
      out[idx] = tre * tre + tim * tim;
    }
  }
}

extern "C" void kernel_launch(void* const* d_in, const int* in_sizes, int n_in,
                              void* d_out, int out_size, void* d_ws, size_t ws_size,
                              hipStream_t stream) {
  const float* E    = (const float*)d_in[0];
  const float* ec_w = (const float*)d_in[1];
  const float* ec_b = (const float*)d_in[2];
  const float* th_w = (const float*)d_in[3];
  const float* th_b = (const float*)d_in[4];
  const float* gr_w = (const float*)d_in[5];
  const float* gr_b = (const float*)d_in[6];
  const float* gl_w = (const float*)d_in[7];
  const float* gl_b = (const float*)d_in[8];
  const float* e0   = (const float*)d_in[9];
  float* out = (float*)d_out;
  float* ws  = (float*)d_ws;

  const int n = in_sizes[0];          // BATCH (4194304)
  setup_params<<<1, 32, 0, stream>>>(ec_w, ec_b, th_w, th_b, gr_w, gr_b,
                                     gl_w, gl_b, e0, ws);

  const int ntiles = (n + 15) >> 4;
  int blocks = (ntiles + 7) / 8;      // 8 waves per 256-thread block
  if (blocks > 2048) blocks = 2048;   // 16k waves grid-stride over tiles
  if (blocks < 1) blocks = 1;
  peaks_wmma_kernel<<<blocks, 256, 0, stream>>>(E, out, ws, n);
}